// SpatialGNNLayer_13597866459873
// MI455X (gfx1250) — compile-verified
//
#include <hip/hip_runtime.h>
#include <hip/hip_bf16.h>

typedef __attribute__((ext_vector_type(2))) float v2f;
typedef __attribute__((ext_vector_type(8))) float v8f;

#define FD    128          // feature dim
#define APAD  260          // ldsA row stride in floats (256 cols + pad; 260%64=4 -> conflict-free b64)
#define BPAD  144          // ldsB row stride in float2 (144%32==16 -> halves land 32 banks apart)
#define LDSA_BYTES (128 * APAD * 4)          // 133120
#define LDSB_BYTES (128 * BPAD * 8)          // 147456
#define LN_EPS 1e-5f

// ---------------- Phase 1: edge scatter (mean numerator + degree) ----------------
__global__ void __launch_bounds__(256)
gnn_scatter_kernel(const float* __restrict__ x, const int* __restrict__ ei,
                   float* __restrict__ agg, float* __restrict__ cnt,
                   int E, int N) {
    int e = blockIdx.x * 8 + (threadIdx.x >> 5);
    if (e >= E) return;
    int lane = threadIdx.x & 31;
    int src  = ei[e];          // row 0 of (2,E)
    int dst  = ei[E + e];      // row 1 of (2,E)
    const float4 v = *(const float4*)&x[(size_t)src * FD + lane * 4];
    float* a = &agg[(size_t)dst * FD + lane * 4];
    atomicAdd(a + 0, v.x);
    atomicAdd(a + 1, v.y);
    atomicAdd(a + 2, v.z);
    atomicAdd(a + 3, v.w);
    if (lane == 0) atomicAdd(&cnt[dst], 1.0f);
}

// ------- Phase 2: fused mean + dual-GEMM (f32 WMMA, K'=256) + bias + LN + ReLU -------
__global__ void __launch_bounds__(256)
gnn_gemm_ln_kernel(const float* __restrict__ x,
                   const float* __restrict__ agg, const float* __restrict__ cnt,
                   const float* __restrict__ Wl, const float* __restrict__ bl,
                   const float* __restrict__ Wr,
                   const float* __restrict__ gamma, const float* __restrict__ beta,
                   float* __restrict__ out, int N) {
    extern __shared__ char smem[];
    float*  ldsA = (float*)smem;                    // [128 rows][APAD]  A' = [mean | x]
    float2* ldsB = (float2*)(smem + LDSA_BYTES);    // [128 kp ][BPAD]  B' = [Wl^T ; Wr^T] as k-pairs

    const int tid  = threadIdx.x;
    const int r0   = blockIdx.x * 128;

    // Stage A' = [mean(row) | x(row)] for 128 rows (clamped at N-1; invalid rows never stored).
    for (int idx = tid; idx < 128 * 128; idx += 256) {
        int row = idx >> 7, col = idx & 127;
        int gr  = min(r0 + row, N - 1);
        float c = cnt[gr];
        float m = agg[(size_t)gr * FD + col] / fmaxf(c, 1.0f);
        ldsA[row * APAD + col]       = m;
        ldsA[row * APAD + 128 + col] = x[(size_t)gr * FD + col];
    }
    // Stage B'[k'][n] = (k'<128 ? Wl[n][k'] : Wr[n][k'-128]) packed as (k', k'+1) float2 pairs.
    for (int idx = tid; idx < 128 * 128; idx += 256) {
        int kp = idx >> 7, n = idx & 127;
        int k2 = kp << 1;
        const float* W = (k2 < 128) ? Wl : Wr;
        int kk = k2 & 127;
        ldsB[kp * BPAD + n] = make_float2(W[n * FD + kk], W[n * FD + kk + 1]);
    }
    __syncthreads();

    const int wave = tid >> 5;        // 0..7 -> 16-row stripe
    const int lane = tid & 31;
    const int half = lane >> 4;       // K-offset half (0 or 1)
    const int lr   = lane & 15;

    const float* aRow = &ldsA[((wave << 4) + lr) * APAD];

    v8f acc[8] = {};                  // 8 n-tiles of 16x16 f32
    for (int k0 = 0; k0 < 256; k0 += 4) {
        v2f a = *(const v2f*)&aRow[k0 + (half << 1)];           // A frag: M=lr, K=k0+2*half..+1
        const float2* bRow = &ldsB[((k0 >> 1) + half) * BPAD + lr];
        #pragma unroll
        for (int nt = 0; nt < 8; ++nt) {
            float2 bv = bRow[nt * 16];                          // B frag: K pair, N=nt*16+lr
            v2f b; b.x = bv.x; b.y = bv.y;
            acc[nt] = __builtin_amdgcn_wmma_f32_16x16x4_f32(
                false, a, false, b, (short)0, acc[nt], false, false);
        }
    }

    // Bias add; cache gamma/beta per n-tile.
    float g[8], be[8];
    #pragma unroll
    for (int nt = 0; nt < 8; ++nt) {
        int n = (nt << 4) + lr;
        float bb = bl[n];
        g[nt] = gamma[n]; be[nt] = beta[n];
        #pragma unroll
        for (int v = 0; v < 8; ++v) acc[nt][v] += bb;
    }

    // LayerNorm: row M = v + 8*half lives in one half-wave -> 16-lane shfl_xor reductions.
    float mu[8], rs[8];
    #pragma unroll
    for (int v = 0; v < 8; ++v) {
        float s = 0.f;
        #pragma unroll
        for (int nt = 0; nt < 8; ++nt) s += acc[nt][v];
        #pragma unroll
        for (int off = 1; off < 16; off <<= 1) s += __shfl_xor(s, off, 16);
        mu[v] = s * (1.0f / 128.0f);
    }
    #pragma unroll
    for (int v = 0; v < 8; ++v) {
        float s = 0.f;
        #pragma unroll
        for (int nt = 0; nt < 8; ++nt) { float d = acc[nt][v] - mu[v]; s += d * d; }
        #pragma unroll
        for (int off = 1; off < 16; off <<= 1) s += __shfl_xor(s, off, 16);
        rs[v] = rsqrtf(s * (1.0f / 128.0f) + LN_EPS);
    }

    // Scale/shift + ReLU + store (guarded for the ragged last block).
    #pragma unroll
    for (int v = 0; v < 8; ++v) {
        int rowg = r0 + (wave << 4) + (half << 3) + v;
        if (rowg < N) {
            #pragma unroll
            for (int nt = 0; nt < 8; ++nt) {
                float val = (acc[nt][v] - mu[v]) * rs[v] * g[nt] + be[nt];
                out[(size_t)rowg * FD + (nt << 4) + lr] = fmaxf(val, 0.0f);
            }
        }
    }
}

extern "C" void kernel_launch(void* const* d_in, const int* in_sizes, int n_in,
                              void* d_out, int out_size, void* d_ws, size_t ws_size,
                              hipStream_t stream) {
    const float* x     = (const float*)d_in[0];
    const int*   ei    = (const int*)  d_in[1];   // (2, E) int32
    const float* Wl    = (const float*)d_in[2];
    const float* bl    = (const float*)d_in[3];
    const float* Wr    = (const float*)d_in[4];
    const float* gamma = (const float*)d_in[5];
    const float* beta  = (const float*)d_in[6];
    float* out = (float*)d_out;

    const int N = in_sizes[0] / FD;
    const int E = in_sizes[1] / 2;

    float* agg = (float*)d_ws;                    // N x 128
    float* cnt = agg + (size_t)N * FD;            // N

    hipMemsetAsync(d_ws, 0, (size_t)N * FD * sizeof(float) + (size_t)N * sizeof(float), stream);

    // Phase 1: one wave per edge.
    int blocks1 = (E + 7) / 8;
    gnn_scatter_kernel<<<blocks1, 256, 0, stream>>>(x, ei, agg, cnt, E, N);

    // Phase 2: 128 rows per block, 8 waves x (16 rows x 128 cols) WMMA stripes.
    int blocks2 = (N + 127) / 128;
    gnn_gemm_ln_kernel<<<blocks2, 256, LDSA_BYTES + LDSB_BYTES, stream>>>(
        x, agg, cnt, Wl, bl, Wr, gamma, beta, out, N);
}